// ChebEncoder_55284819034171
// MI455X (gfx1250) — compile-verified
//
#include <hip/hip_runtime.h>

typedef float v2f __attribute__((ext_vector_type(2)));
typedef float v8f __attribute__((ext_vector_type(8)));

#define N_NODES 30000
#define IN_CH   128
#define HID_CH  256
#define N_EDGES 480000

// ---------------- elementwise helpers ----------------

__global__ void zero_kernel(float* __restrict__ p, int n) {
    int i = blockIdx.x * blockDim.x + threadIdx.x;
    for (; i < n; i += gridDim.x * blockDim.x) p[i] = 0.0f;
}

__global__ void negcopy_kernel(const float* __restrict__ in, float* __restrict__ out, int n) {
    int i = blockIdx.x * blockDim.x + threadIdx.x;
    for (; i < n; i += gridDim.x * blockDim.x) out[i] = -in[i];
}

// ---------------- graph normalization ----------------

__global__ void deg_kernel(const long long* __restrict__ ei, float* __restrict__ deg) {
    int e = blockIdx.x * blockDim.x + threadIdx.x;
    if (e >= N_EDGES) return;
    long long s = ei[e];
    long long d = ei[N_EDGES + e];
    if (s != d) atomicAdd(&deg[(int)s], 1.0f);
}

__global__ void dinv_kernel(float* __restrict__ deg, int n) {
    int i = blockIdx.x * blockDim.x + threadIdx.x;
    if (i >= n) return;
    float d = deg[i];
    deg[i] = (d > 0.0f) ? rsqrtf(d) : 0.0f;
}

__global__ void edgew_kernel(const long long* __restrict__ ei,
                             const float* __restrict__ dinv,
                             float* __restrict__ w) {
    int e = blockIdx.x * blockDim.x + threadIdx.x;
    if (e >= N_EDGES) return;
    long long s = ei[e];
    long long d = ei[N_EDGES + e];
    w[e] = (s != d) ? (-dinv[(int)s] * dinv[(int)d]) : 0.0f;
}

// ---------------- sparse scatter: out[dst] += alpha * w[e] * in[src] ----------------
// one block per edge, one thread per channel (C = blockDim.x)

__global__ void spmm_kernel(const long long* __restrict__ ei,
                            const float* __restrict__ w,
                            const float* __restrict__ in,
                            float* __restrict__ out,
                            float alpha, int C) {
    int e = blockIdx.x;
    float we = w[e];
    if (we == 0.0f) return;
    int s = (int)ei[e];
    int d = (int)ei[N_EDGES + e];
    int c = threadIdx.x;
    float v = alpha * we * in[(size_t)s * C + c];
    atomicAdd(&out[(size_t)d * C + c], v);
}

// ---------------- fused ChebConv GEMM via V_WMMA_F32_16X16X4_F32 ----------------
// out[N, OUTC] = A0@W[0] + A1@W[1] + A2@W[2] + bias   (optional ReLU)
// One wave per 16x16 output tile; blockDim=(32,4): 4 row-tiles per block, shared col0.
// W panel for the block's 16 columns is staged in LDS as interleaved (k,k+1) pairs
// so each lane's B operand is one 8B-aligned ds_load_b64 (conflict-free across 64 banks).

template <int INC, int OUTC, int RELU>
__global__ void __launch_bounds__(128)
cheb_gemm_kernel(const float* __restrict__ A0,
                 const float* __restrict__ A1,
                 const float* __restrict__ A2,
                 const float* __restrict__ W,      // [3, INC, OUTC] row-major
                 const float* __restrict__ bias,   // [OUTC]
                 float* __restrict__ out) {        // [N, OUTC]
    // LDS layout: lw[((t*(INC/2) + k/2)*16 + c)*2 + (k&1)] = W[t][k][col0+c]
    __shared__ float lw[3 * INC * 16];

    const int lane  = threadIdx.x;                 // 0..31 (wave32)
    const int col0  = blockIdx.x * 16;
    const int mtile = blockIdx.y * 4 + threadIdx.y;   // wave-uniform
    const bool active = (mtile * 16 < N_NODES);

    // ---- stage W panel (all 128 threads participate before the barrier) ----
    {
        const int tid = threadIdx.y * 32 + lane;
        #pragma unroll 4
        for (int i = tid; i < 3 * INC * 16; i += 128) {
            const int t = i / (INC * 16);
            const int r = i - t * (INC * 16);
            const int k = r >> 4;
            const int c = r & 15;
            lw[((t * (INC / 2) + (k >> 1)) * 16 + c) * 2 + (k & 1)] =
                W[(size_t)t * INC * OUTC + (size_t)k * OUTC + col0 + c];
        }
    }
    __syncthreads();
    if (!active) return;                // uniform per wave: EXEC all-1s for WMMA

    const int row0 = mtile * 16;
    const int half = lane >> 4;         // 0: K={0,1}/rows 0-7 ; 1: K={2,3}/rows 8-15
    const int mr   = lane & 15;

    v8f acc = {};

    const float* As[3] = {A0, A1, A2};
    #pragma unroll
    for (int t = 0; t < 3; ++t) {
        const float* __restrict__ arow = As[t] + (size_t)(row0 + mr) * INC;
        const float* __restrict__ lwt  = lw + (t * (INC / 2)) * 32;
        #pragma unroll 4
        for (int k = 0; k < INC; k += 4) {
            const int ka = k + 2 * half;
            // A tile 16x4: lane holds A[row0+mr][ka..ka+1]  (global b64)
            v2f a = *(const v2f*)(arow + ka);
            // B tile 4x16: lane holds W[ka..ka+1][col0+mr]  (lds b64)
            v2f b = *(const v2f*)(lwt + ((ka >> 1) * 16 + mr) * 2);
            acc = __builtin_amdgcn_wmma_f32_16x16x4_f32(
                false, a, false, b, (short)0, acc, false, false);
        }
    }

    const float bv = bias[col0 + mr];
    #pragma unroll
    for (int r = 0; r < 8; ++r) {
        // C/D layout: VGPR r -> row r (lanes 0-15) / r+8 (lanes 16-31), col = lane&15
        float v = acc[r] + bv;
        if (RELU) v = fmaxf(v, 0.0f);
        out[(size_t)(row0 + r + 8 * half) * OUTC + col0 + mr] = v;
    }
}

// ---------------- launch ----------------

extern "C" void kernel_launch(void* const* d_in, const int* in_sizes, int n_in,
                              void* d_out, int out_size, void* d_ws, size_t ws_size,
                              hipStream_t stream) {
    const float*     x  = (const float*)d_in[0];           // [N, 128]
    const long long* ei = (const long long*)d_in[1];       // [2, E] int64
    const float*     W1 = (const float*)d_in[2];           // [3, 128, 256]
    const float*     b1 = (const float*)d_in[3];           // [256]
    const float*     W2 = (const float*)d_in[4];           // [3, 256, 128]
    const float*     b2 = (const float*)d_in[5];           // [128]
    float* out = (float*)d_out;                            // [N, 128]

    // workspace layout (floats)
    float* ws   = (float*)d_ws;
    size_t off  = 0;
    float* deg  = ws + off; off += N_NODES;                       // degree -> deg^-1/2
    float* w    = ws + off; off += N_EDGES;                       // edge weights
    float* h    = ws + off; off += (size_t)N_NODES * HID_CH;      // hidden activations
    float* R1   = ws + off; off += (size_t)N_NODES * HID_CH;      // Tx1a|Tx2a, later Tx2b
    float* R2   = ws + off; off += (size_t)N_NODES * HID_CH;      // Tx1b

    const int T256 = 256;
    // --- normalization ---
    zero_kernel<<<(N_NODES + T256 - 1) / T256, T256, 0, stream>>>(deg, N_NODES);
    deg_kernel<<<(N_EDGES + T256 - 1) / T256, T256, 0, stream>>>(ei, deg);
    dinv_kernel<<<(N_NODES + T256 - 1) / T256, T256, 0, stream>>>(deg, N_NODES);
    edgew_kernel<<<(N_EDGES + T256 - 1) / T256, T256, 0, stream>>>(ei, deg, w);

    // --- layer 1 (C = 128) ---
    float* Tx1a = R1;
    float* Tx2a = R1 + (size_t)N_NODES * IN_CH;
    int n128 = N_NODES * IN_CH;
    zero_kernel<<<4096, T256, 0, stream>>>(Tx1a, n128);
    spmm_kernel<<<N_EDGES, IN_CH, 0, stream>>>(ei, w, x, Tx1a, 1.0f, IN_CH);
    negcopy_kernel<<<4096, T256, 0, stream>>>(x, Tx2a, n128);
    spmm_kernel<<<N_EDGES, IN_CH, 0, stream>>>(ei, w, Tx1a, Tx2a, 2.0f, IN_CH);

    dim3 blk(32, 4);
    dim3 g1(HID_CH / 16, (N_NODES / 16 + 3) / 4);   // 16 x 469
    cheb_gemm_kernel<IN_CH, HID_CH, 1><<<g1, blk, 0, stream>>>(
        x, Tx1a, Tx2a, W1, b1, h);

    // --- layer 2 (C = 256) ---
    float* Tx1b = R2;
    float* Tx2b = R1;   // Tx1a/Tx2a dead after GEMM1
    int n256 = N_NODES * HID_CH;
    zero_kernel<<<8192, T256, 0, stream>>>(Tx1b, n256);
    spmm_kernel<<<N_EDGES, HID_CH, 0, stream>>>(ei, w, h, Tx1b, 1.0f, HID_CH);
    negcopy_kernel<<<8192, T256, 0, stream>>>(h, Tx2b, n256);
    spmm_kernel<<<N_EDGES, HID_CH, 0, stream>>>(ei, w, Tx1b, Tx2b, 2.0f, HID_CH);

    dim3 g2(IN_CH / 16, (N_NODES / 16 + 3) / 4);    // 8 x 469
    cheb_gemm_kernel<HID_CH, IN_CH, 0><<<g2, blk, 0, stream>>>(
        h, Tx1b, Tx2b, W2, b2, out);
}